// FeatureTransformerSlice_3307124818497
// MI455X (gfx1250) — compile-verified
//
#include <hip/hip_runtime.h>
#include <hip/hip_bf16.h>

// FeatureTransformerSlice forward: out[b] = bias + sum_k val[b,k] * W[idx[b,k], :]
//   B = 16384, K = 32 active features/row, O = 1024, all f32.
//
// Roofline: 1.07 GFLOP vs 2.15 GB of gathered weight reads (AI ~ 0.5 FLOP/B)
// -> pure bandwidth problem. Weight table = 176 MiB, resident in the 192 MB L2,
// so steady state is an L2-bandwidth gather; HBM floor would be ~95 us at
// 23.3 TB/s. No dense-GEMM structure exists (the B operand W[idx[b,k],:]
// varies per output row), so WMMA cannot reduce data movement — the CDNA5
// features that apply are the async-memory path and cache-policy control:
//   * idx/val are wave-uniform per row -> one GLOBAL_LOAD_ASYNC_TO_LDS_B128
//     DMA stages them (ASYNCcnt), readfirstlane scalarizes them, and the
//     weight gather becomes saddr-form coalesced GLOBAL_LOAD_B128s with all
//     address math in SALU.
//   * output (64 MiB, write-once) uses nontemporal B128 stores so it cannot
//     evict the L2-resident weights.
//   * unroll 16 -> 16 B128 gathers in flight per wave for deep MLP.

#define FT_K       32     // MAX_ACTIVE
#define FT_O       1024   // OUTPUT_SIZE
#define FT_THREADS 256    // 8 wave32; thread t owns outputs [4t, 4t+4)

typedef float v4f __attribute__((ext_vector_type(4)));

__global__ __launch_bounds__(FT_THREADS) void ft_fwd_kernel(
    const int*   __restrict__ fidx,    // [B, 32]
    const float* __restrict__ fval,    // [B, 32]
    const float* __restrict__ weight,  // [45056, 1024]
    const float* __restrict__ bias,    // [1024]
    float*       __restrict__ out)     // [B, 1024]
{
    // 128 B of indices followed by 128 B of value bits for this batch row.
    __shared__ __align__(16) int s_feat[2 * FT_K];

    const int b = blockIdx.x;
    const int t = threadIdx.x;

    // ---- Stage idx/val for this row into LDS via one async DMA -------------
    // Lanes 0..7 copy the 32 indices (16 B each), lanes 8..15 the 32 values.
    // Branchless source select keeps wave 0 convergent.
    if (t < 16) {
        const unsigned long long gi =
            (unsigned long long)(size_t)(fidx + (size_t)b * FT_K);
        const unsigned long long gv =
            (unsigned long long)(size_t)(fval + (size_t)b * FT_K);
        unsigned long long g = ((t & 8) ? gv : gi) + (unsigned long long)(t & 7) * 16ull;
        // Low 32 bits of a generic shared-aperture address == LDS byte offset.
        unsigned lds = (unsigned)(size_t)(&s_feat[0]) + (unsigned)t * 16u;
        asm volatile("global_load_async_to_lds_b128 %0, %1, off"
                     :: "v"(lds), "v"(g) : "memory");
    }
    asm volatile("s_wait_asynccnt 0x0" ::: "memory");  // wave 0: DMA done; others: no-op
    __syncthreads();

    // ---- Accumulate: 4 consecutive outputs per thread ----------------------
    const int col = t * 4;                 // thread-constant output columns
    v4f acc = *(const v4f*)(bias + col);   // init with bias

    const float* __restrict__ wcol = weight + col;

#pragma unroll 16
    for (int k = 0; k < FT_K; ++k) {
        // Row-uniform -> force into SGPRs: SALU address math + saddr B128 loads.
        int raw = __builtin_amdgcn_readfirstlane(s_feat[k]);
        int neg = raw >> 31;                         // all-ones for empty (-1) slots
        int ik  = raw & ~neg;                        // safe index (0 if empty)
        unsigned vb = (unsigned)__builtin_amdgcn_readfirstlane(s_feat[FT_K + k]);
        vb &= ~(unsigned)neg;                        // value -> +0.0f if empty
        const float vk = __uint_as_float(vb);

        const v4f w = *(const v4f*)(wcol + (size_t)ik * FT_O);  // coalesced B128
        acc.x = fmaf(vk, w.x, acc.x);
        acc.y = fmaf(vk, w.y, acc.y);
        acc.z = fmaf(vk, w.z, acc.z);
        acc.w = fmaf(vk, w.w, acc.w);
    }

    // Streamed once, never re-read: nontemporal so the output stream cannot
    // evict the L2-resident weight table.
    __builtin_nontemporal_store(acc, (v4f*)(out + (size_t)b * FT_O + col));
}

extern "C" void kernel_launch(void* const* d_in, const int* in_sizes, int n_in,
                              void* d_out, int out_size, void* d_ws, size_t ws_size,
                              hipStream_t stream) {
    const int*   fidx   = (const int*)  d_in[0];  // feature_indices [B,32] i32
    const float* fval   = (const float*)d_in[1];  // feature_values  [B,32] f32
    const float* weight = (const float*)d_in[2];  // [45056,1024] f32
    const float* bias   = (const float*)d_in[3];  // [1024] f32
    float*       out    = (float*)d_out;          // [B,1024] f32

    const int batch = in_sizes[0] / FT_K;         // 16384
    ft_fwd_kernel<<<dim3(batch), dim3(FT_THREADS), 0, stream>>>(fidx, fval, weight, bias, out);
}